// quantizationLayer_49563922596433
// MI455X (gfx1250) — compile-verified
//
#include <hip/hip_runtime.h>

// out[i] = sum_{k=0..6} a[k] * tanh(5*(x[i] + b[k]))
// x: (2048,4096) f32, weight: (7,2) f32 with weight[k][0]=a[k], weight[k][1]=b[k].
// Memory-bound streaming kernel: 64 MB total traffic -> ~2.75 us floor @ 23.3 TB/s.
// Uses gfx1250 native v_tanh_f32 so the transcendental pipe stays under the HBM roof.

typedef __attribute__((ext_vector_type(4))) float v4f;

#define MAGIC_C 5.0f
#define KTERMS 7

__device__ __forceinline__ float dev_tanh(float y) {
#if __has_builtin(__builtin_amdgcn_tanhf)
  return __builtin_amdgcn_tanhf(y);   // gfx1250: v_tanh_f32 (1 trans op)
#else
  return tanhf(y);                    // fallback if toolchain lacks the builtin
#endif
}

__global__ __launch_bounds__(256) void tanh_feature_mix_kernel(
    const float* __restrict__ x,
    const float* __restrict__ weight,   // (7,2) row-major
    float* __restrict__ out,
    int n)
{
  // Broadcast coefficients: uniform-address loads -> scalar loads, cached.
  float a[KTERMS], c[KTERMS];
#pragma unroll
  for (int k = 0; k < KTERMS; ++k) {
    a[k] = weight[2 * k];
    c[k] = MAGIC_C * weight[2 * k + 1];   // arg = fma(5, x, 5*b[k])
  }

  const int i = blockIdx.x * blockDim.x + threadIdx.x;
  const long long base = (long long)i * 4;

  if (base + 4 <= (long long)n) {
    // 128-bit non-temporal load: streamed once, no reuse -> don't pollute L2.
    v4f v = __builtin_nontemporal_load((const v4f*)(x + base));
    float s0 = 0.f, s1 = 0.f, s2 = 0.f, s3 = 0.f;
#pragma unroll
    for (int k = 0; k < KTERMS; ++k) {
      s0 = __builtin_fmaf(a[k], dev_tanh(__builtin_fmaf(MAGIC_C, v.x, c[k])), s0);
      s1 = __builtin_fmaf(a[k], dev_tanh(__builtin_fmaf(MAGIC_C, v.y, c[k])), s1);
      s2 = __builtin_fmaf(a[k], dev_tanh(__builtin_fmaf(MAGIC_C, v.z, c[k])), s2);
      s3 = __builtin_fmaf(a[k], dev_tanh(__builtin_fmaf(MAGIC_C, v.w, c[k])), s3);
    }
    v4f r = {s0, s1, s2, s3};
    __builtin_nontemporal_store(r, (v4f*)(out + base));
  } else if (base < (long long)n) {
    // Tail (not hit for 2048*4096, but keep the kernel shape-robust).
    for (long long j = base; j < (long long)n; ++j) {
      float xv = x[j], s = 0.f;
#pragma unroll
      for (int k = 0; k < KTERMS; ++k)
        s = __builtin_fmaf(a[k], dev_tanh(__builtin_fmaf(MAGIC_C, xv, c[k])), s);
      out[j] = s;
    }
  }
}

extern "C" void kernel_launch(void* const* d_in, const int* in_sizes, int n_in,
                              void* d_out, int out_size, void* d_ws, size_t ws_size,
                              hipStream_t stream) {
  const float* x = (const float*)d_in[0];      // 2048*4096 f32
  const float* w = (const float*)d_in[1];      // 7*2 f32
  float* out = (float*)d_out;                  // 2048*4096 f32
  const int n = in_sizes[0];

  const int nv4 = (n + 3) / 4;                 // one float4 per thread
  const int threads = 256;                     // 8 wave32 per block
  const int blocks = (nv4 + threads - 1) / threads;

  tanh_feature_mix_kernel<<<blocks, threads, 0, stream>>>(x, w, out, n);
}